// GCN_11209864642750
// MI455X (gfx1250) — compile-verified
//
#include <hip/hip_runtime.h>

typedef float v2f __attribute__((ext_vector_type(2)));
typedef float v8f __attribute__((ext_vector_type(8)));

#define IN_DIM 21
#define IN_PAD 24

// ---------------- degree / normalization ----------------

__global__ void k_set1(float* __restrict__ p, int n) {
  int i = blockIdx.x * blockDim.x + threadIdx.x;
  if (i < n) p[i] = 1.0f;  // self-loop contributes 1 to every degree
}

__global__ void k_deg(const int* __restrict__ dst, float* __restrict__ deg, int e) {
  int i = blockIdx.x * blockDim.x + threadIdx.x;
  if (i < e) atomicAdd(&deg[dst[i]], 1.0f);
}

__global__ void k_rsqrt(const float* __restrict__ deg, float* __restrict__ dinv, int n) {
  int i = blockIdx.x * blockDim.x + threadIdx.x;
  if (i < n) dinv[i] = rsqrtf(deg[i]);  // deg >= 1 always
}

// Zero-pad X (rows x fin) into Xp (rows x fpad): pad trailing columns.
__global__ void k_pad_cols(const float* __restrict__ X, float* __restrict__ Xp,
                           int rows, int fin, int fpad) {
  int i = blockIdx.x * blockDim.x + threadIdx.x;
  if (i >= rows * fpad) return;
  int row = i / fpad;
  int c = i - row * fpad;
  Xp[i] = (c < fin) ? X[(size_t)row * fin + c] : 0.0f;
}

// Zero-pad W (kin x fout) into Wp (kpad x fout): pad trailing rows.
__global__ void k_pad_rows(const float* __restrict__ W, float* __restrict__ Wp,
                           int kin, int kpad, int fout) {
  int i = blockIdx.x * blockDim.x + threadIdx.x;
  if (i >= kpad * fout) return;
  int row = i / fout;
  Wp[i] = (row < kin) ? W[i] : 0.0f;
}

// ---------------- fp32 WMMA GEMM: H = X * W ----------------
// One wave = one 16(M) x 16(N) fp32 tile via V_WMMA_F32_16X16X4_F32.
// Requires fin % 4 == 0 (caller pads). fout % 16 == 0.
// A layout (16x4): lanes 0-15 hold K=k0,k0+1; lanes 16-31 hold K=k0+2,k0+3.
// B layout (4x16): lane&15 = column; lane>>4 selects K pair.
// C/D layout: vgpr r, lanes 0-15 -> row r; lanes 16-31 -> row r+8.
__global__ __launch_bounds__(32)
void k_gemm_wmma(const float* __restrict__ X, const float* __restrict__ W,
                 float* __restrict__ H, int n, int fin, int fout) {
  const int lane = threadIdx.x;
  const int half = lane >> 4;   // 0 or 1
  const int lid  = lane & 15;
  const int mbase = blockIdx.x * 16;
  const int nbase = blockIdx.y * 16;

  int rowA = mbase + lid;
  int rA = rowA < n ? rowA : n - 1;  // clamp: rows >= n never stored
  int col = nbase + lid;

  const float* pa = X + (size_t)rA * fin + half * 2;       // 8B aligned
  const float* pb = W + (size_t)(half * 2) * fout + col;
  const size_t bstep = (size_t)4 * fout;

  v8f acc = {};
  for (int k0 = 0; k0 < fin; k0 += 4) {
    v2f a = *(const v2f*)pa;           // K pair: single b64 load
    v2f bm;
    bm.x = pb[0];
    bm.y = pb[fout];
    acc = __builtin_amdgcn_wmma_f32_16x16x4_f32(false, a, false, bm,
                                                (short)0, acc, false, false);
    pa += 4;
    pb += bstep;
  }

  float* po = H + (size_t)(mbase + half * 8) * fout + col;
  if (mbase + 16 <= n) {               // full tile: unguarded stores
#pragma unroll
    for (int r = 0; r < 8; ++r) po[(size_t)r * fout] = acc[r];
  } else {
#pragma unroll
    for (int r = 0; r < 8; ++r)
      if (mbase + half * 8 + r < n) po[(size_t)r * fout] = acc[r];
  }
}

// ---------------- aggregation ----------------

// out = bias + h * dinv^2   (self-loop term + bias, fused init)
__global__ void k_agg_init(const float* __restrict__ h, const float* __restrict__ dinv,
                           const float* __restrict__ bias, float* __restrict__ out,
                           int total, int fshift, int fmask) {
  int i = blockIdx.x * blockDim.x + threadIdx.x;
  if (i >= total) return;
  int node = i >> fshift;
  int c = i & fmask;
  float di = dinv[node];
  out[i] = bias[c] + h[i] * di * di;
}

// out[dst] += h[src] * dinv[src] * dinv[dst]; (edge, 4-float chunk) per thread.
// Consecutive lanes cover one edge's feature row -> coalesced 16B gathers.
__global__ void k_agg_edges(const float* __restrict__ h, float* __restrict__ out,
                            const int* __restrict__ src, const int* __restrict__ dst,
                            const float* __restrict__ dinv,
                            long long total, int fshift, int cshift) {
  long long tid = (long long)blockIdx.x * blockDim.x + threadIdx.x;
  if (tid >= total) return;
  int e = (int)(tid >> cshift);
  int c = ((int)tid & ((1 << cshift) - 1)) << 2;
  int s = src[e];
  int d = dst[e];
  float w = dinv[s] * dinv[d];
  const float4 hv = *(const float4*)(h + ((size_t)s << fshift) + c);
  float* o = out + ((size_t)d << fshift) + c;
  atomicAdd(o + 0, hv.x * w);
  atomicAdd(o + 1, hv.y * w);
  atomicAdd(o + 2, hv.z * w);
  atomicAdd(o + 3, hv.w * w);
}

__global__ void k_relu(float* __restrict__ p, int total) {
  int i = blockIdx.x * blockDim.x + threadIdx.x;
  if (i < total) p[i] = fmaxf(p[i], 0.0f);
}

// ---------------- MLP head: relu(h@Wl1+bl1) @ Wl2 + bl2 ----------------
__global__ void k_mlp(const float* __restrict__ h,
                      const float* __restrict__ Wl1, const float* __restrict__ bl1,
                      const float* __restrict__ Wl2, const float* __restrict__ bl2,
                      float* __restrict__ out, int n) {
  int i = blockIdx.x * blockDim.x + threadIdx.x;
  if (i >= n) return;
  float hv[64];
#pragma unroll
  for (int k = 0; k < 64; ++k) hv[k] = h[(size_t)i * 64 + k];
  float o = bl2[0];
  for (int j = 0; j < 20; ++j) {
    float s = bl1[j];
#pragma unroll 8
    for (int k = 0; k < 64; ++k) s += hv[k] * Wl1[k * 20 + j];  // uniform -> SMEM
    s = fmaxf(s, 0.0f);
    o += s * Wl2[j];
  }
  out[i] = o;
}

// ---------------- launcher ----------------

extern "C" void kernel_launch(void* const* d_in, const int* in_sizes, int n_in,
                              void* d_out, int out_size, void* d_ws, size_t ws_size,
                              hipStream_t stream) {
  const float* x   = (const float*)d_in[0];
  const int*   ei  = (const int*)d_in[1];
  const float* W1  = (const float*)d_in[2];
  const float* b1  = (const float*)d_in[3];
  const float* W2  = (const float*)d_in[4];
  const float* b2  = (const float*)d_in[5];
  const float* W3  = (const float*)d_in[6];
  const float* b3  = (const float*)d_in[7];
  const float* Wl1 = (const float*)d_in[8];
  const float* bl1 = (const float*)d_in[9];
  const float* Wl2 = (const float*)d_in[10];
  const float* bl2 = (const float*)d_in[11];

  const int n = in_sizes[0] / IN_DIM;  // 100000
  const int E = in_sizes[1] / 2;       // 3200000
  const int* src = ei;
  const int* dst = ei + E;
  float* out = (float*)d_out;

  const size_t bufBytes = (size_t)n * 128 * sizeof(float);
  char* ws = (char*)d_ws;
  float* bufA = (float*)ws;                    // GEMM output h = x@W
  float* bufB = (float*)(ws + bufBytes);       // aggregated output
  float* deg  = (float*)(ws + 2 * bufBytes);
  float* dinv = deg + n;
  float* bufP = dinv + n;                      // padded layer-1 input (n x 24)
  float* W1p  = bufP + (size_t)n * IN_PAD;     // padded W1 (24 x 64)

  const dim3 B(256);
  k_set1 <<<(n + 255) / 256, B, 0, stream>>>(deg, n);
  k_deg  <<<(E + 255) / 256, B, 0, stream>>>(dst, deg, E);
  k_rsqrt<<<(n + 255) / 256, B, 0, stream>>>(deg, dinv, n);
  k_pad_cols<<<(n * IN_PAD + 255) / 256, B, 0, stream>>>(x, bufP, n, IN_DIM, IN_PAD);
  k_pad_rows<<<(IN_PAD * 64 + 255) / 256, B, 0, stream>>>(W1, W1p, IN_DIM, IN_PAD, 64);

  auto layer = [&](const float* in, const float* Wt, const float* bias,
                   int fin, int fout, int fshift, int cshift) {
    dim3 g((n + 15) / 16, fout / 16);
    k_gemm_wmma<<<g, dim3(32), 0, stream>>>(in, Wt, bufA, n, fin, fout);
    int tot = n * fout;
    k_agg_init<<<(tot + 255) / 256, B, 0, stream>>>(bufA, dinv, bias, bufB,
                                                    tot, fshift, fout - 1);
    long long etot = (long long)E << cshift;
    unsigned eblocks = (unsigned)((etot + 255) / 256);
    k_agg_edges<<<eblocks, B, 0, stream>>>(bufA, bufB, src, dst, dinv,
                                           etot, fshift, cshift);
    k_relu<<<(tot + 255) / 256, B, 0, stream>>>(bufB, tot);
  };

  layer(bufP, W1p, b1, IN_PAD,  64, 6, 4);
  layer(bufB, W2,  b2,  64,    128, 7, 5);
  layer(bufB, W3,  b3, 128,     64, 6, 4);

  k_mlp<<<(n + 255) / 256, B, 0, stream>>>(bufB, Wl1, bl1, Wl2, bl2, out, n);
}